// GCN_71854802862599
// MI455X (gfx1250) — compile-verified
//
#include <hip/hip_runtime.h>

typedef float v2f __attribute__((ext_vector_type(2)));
typedef float v8f __attribute__((ext_vector_type(8)));

// ---------------------------------------------------------------------------
// Degree / norm kernels
// ---------------------------------------------------------------------------
__global__ void k_init_deg(float* deg, int n) {
    int i = blockIdx.x * blockDim.x + threadIdx.x;
    if (i < n) deg[i] = 1.0f;  // self-loop contributes 1 to every node's degree
}

__global__ void k_deg_edges(const long long* __restrict__ dst, float* __restrict__ deg, int e) {
    int i = blockIdx.x * blockDim.x + threadIdx.x;
    if (i < e) atomicAdd(&deg[(int)dst[i]], 1.0f);
}

__global__ void k_rsqrt(float* deg, int n) {
    int i = blockIdx.x * blockDim.x + threadIdx.x;
    if (i < n) deg[i] = rsqrtf(deg[i]);  // deg >= 1, in-place -> dinv
}

// ---------------------------------------------------------------------------
// WMMA GEMM: C[N,NCOLS] = A[N,64] @ B[64,NCOLS], NCOLS in {32,64}, rows%16==0
//
// Block = 128 threads (4 waves). B (<=16KB) is staged into LDS once per block
// via gfx1250 async copies (global_load_async_to_lds_b128, ASYNCcnt), then the
// block grid-strides over 16-row tiles; wave w owns column tile w.
//
// Per 16x16 tile: 16 chained V_WMMA_F32_16X16X4_F32 over K=64.
//   A frag (16x4 f32): m = lane&15, kh = lane>>4; a[j] = A[m][k0 + 2*kh + j]
//   B frag (4x16 f32): b[j] = B[k0 + 2*kh + j][n], n = lane&15   (from LDS)
//   D (16x16 f32):     acc[r] -> C[r + 8*kh][n]
// ---------------------------------------------------------------------------
template <int NCOLS>
__global__ void k_gemm_wmma(const float* __restrict__ A,
                            const float* __restrict__ B,
                            float* __restrict__ C, int ntiles) {
    __shared__ float sB[64 * NCOLS];

    // ---- stage B into LDS with async copies (16B per op, 128 threads) ----
    for (int i = threadIdx.x * 4; i < 64 * NCOLS; i += 128 * 4) {
        unsigned lds_addr = (unsigned)(size_t)(const void*)&sB[i];
        const float* gptr = B + i;
        asm volatile("global_load_async_to_lds_b128 %0, %1, off"
                     :: "v"(lds_addr), "v"(gptr) : "memory");
    }
    asm volatile("s_wait_asynccnt 0x0" ::: "memory");
    __syncthreads();

    const int lane = threadIdx.x & 31;
    const int ct   = threadIdx.x >> 5;  // column tile = wave id
    if (ct * 16 >= NCOLS) return;       // (NCOLS==32: waves 2,3 idle after stage)

    const int m    = lane & 15;
    const int kh   = lane >> 4;
    const int ncol = ct * 16 + m;

    for (int mt = blockIdx.x; mt < ntiles; mt += gridDim.x) {
        const float* __restrict__ arow = A + ((size_t)(mt * 16 + m)) * 64;
        v8f acc = {};
#pragma unroll
        for (int k0 = 0; k0 < 64; k0 += 4) {
            v2f a, b;
            a.x = arow[k0 + 2 * kh + 0];
            a.y = arow[k0 + 2 * kh + 1];
            b.x = sB[(k0 + 2 * kh + 0) * NCOLS + ncol];
            b.y = sB[(k0 + 2 * kh + 1) * NCOLS + ncol];
            acc = __builtin_amdgcn_wmma_f32_16x16x4_f32(
                /*neg_a=*/false, a, /*neg_b=*/false, b,
                /*c_mod=*/(short)0, acc, /*reuse_a=*/false, /*reuse_b=*/false);
        }
#pragma unroll
        for (int r = 0; r < 8; ++r) {
            int row = mt * 16 + r + 8 * kh;
            C[(size_t)row * NCOLS + ncol] = acc[r];
        }
    }
}

// ---------------------------------------------------------------------------
// Layer-1 aggregation
// ---------------------------------------------------------------------------
// agg[i][:] = dinv[i]^2 * h1[i][:]   (self-loop message; also zero-init)
__global__ void k_self1(const float* __restrict__ h1, const float* __restrict__ dinv,
                        float* __restrict__ agg, int n64) {
    int i = blockIdx.x * blockDim.x + threadIdx.x;
    if (i < n64) {
        float d = dinv[i >> 6];
        agg[i] = d * d * h1[i];
    }
}

// wave-per-edge scatter: agg[dst] += norm * h1[src]; lane handles dims 2l,2l+1
__global__ void k_edge1(const long long* __restrict__ src, const long long* __restrict__ dst,
                        const float* __restrict__ dinv, const float* __restrict__ h1,
                        float* __restrict__ agg, int e) {
    int t = blockIdx.x * blockDim.x + threadIdx.x;
    int eid = t >> 5, lane = t & 31;
    if (eid >= e) return;
    int s = (int)src[eid], d = (int)dst[eid];
    float w = dinv[s] * dinv[d];
    const v2f hv = *(const v2f*)(h1 + (size_t)s * 64 + 2 * lane);  // one b64 gather
    float* ad = agg + (size_t)d * 64 + 2 * lane;
    atomicAdd(&ad[0], w * hv.x);
    atomicAdd(&ad[1], w * hv.y);
}

__global__ void k_bias_relu(const float* __restrict__ agg, const float* __restrict__ b1,
                            float* __restrict__ h, int n64) {
    int i = blockIdx.x * blockDim.x + threadIdx.x;
    if (i < n64) {
        float v = agg[i] + b1[i & 63];
        h[i] = v > 0.0f ? v : 0.0f;
    }
}

// ---------------------------------------------------------------------------
// Layer-2: pad W2 (64x19 -> 64x32), then aggregate into packed [N,19] output
// ---------------------------------------------------------------------------
__global__ void k_pad_w2(const float* __restrict__ W2, float* __restrict__ W2p, int c2) {
    int i = blockIdx.x * blockDim.x + threadIdx.x;  // 64*32 threads
    if (i < 64 * 32) {
        int k = i >> 5, n = i & 31;
        W2p[i] = (n < c2) ? W2[k * c2 + n] : 0.0f;
    }
}

// out[i][d] = dinv[i]^2 * h2[i][d] + b2[d]   (self-loop + bias; zero-init of out)
__global__ void k_self2(const float* __restrict__ h2, const float* __restrict__ dinv,
                        const float* __restrict__ b2, float* __restrict__ out,
                        int n, int c2) {
    int t = blockIdx.x * blockDim.x + threadIdx.x;  // n*32 threads
    int node = t >> 5, d = t & 31;
    if (node < n && d < c2) {
        float di = dinv[node];
        out[(size_t)node * c2 + d] = di * di * h2[(size_t)node * 32 + d] + b2[d];
    }
}

__global__ void k_edge2(const long long* __restrict__ src, const long long* __restrict__ dst,
                        const float* __restrict__ dinv, const float* __restrict__ h2,
                        float* __restrict__ out, int e, int c2) {
    int t = blockIdx.x * blockDim.x + threadIdx.x;
    int eid = t >> 5, d = t & 31;
    if (eid >= e || d >= c2) return;
    int s = (int)src[eid], dn = (int)dst[eid];
    float w = dinv[s] * dinv[dn];
    atomicAdd(&out[(size_t)dn * c2 + d], w * h2[(size_t)s * 32 + d]);
}

// ---------------------------------------------------------------------------
// Host-side orchestration (all launches on `stream`; graph-capture safe)
// ---------------------------------------------------------------------------
extern "C" void kernel_launch(void* const* d_in, const int* in_sizes, int n_in,
                              void* d_out, int out_size, void* d_ws, size_t ws_size,
                              hipStream_t stream) {
    const float*     x   = (const float*)d_in[0];
    const long long* ei  = (const long long*)d_in[1];  // int64 [2, E]
    const float*     W1  = (const float*)d_in[2];
    const float*     b1  = (const float*)d_in[3];
    const float*     W2  = (const float*)d_in[4];
    const float*     b2  = (const float*)d_in[5];
    float*           out = (float*)d_out;

    const int N  = in_sizes[0] / 64;          // 100000 (multiple of 16)
    const int E  = in_sizes[1] / 2;           // 1600000
    const int C2 = in_sizes[5];               // 19

    const long long* srcI = ei;
    const long long* dstI = ei + E;

    // workspace layout (floats): [dinv N][h1/h N*64][agg/h2 N*64][W2p 2048]
    float* ws   = (float*)d_ws;
    float* dinv = ws;
    float* h1   = ws + N;                              // x@W1, later relu'd h
    float* agg  = ws + N + (size_t)N * 64;             // layer-1 acc, later h2 [N,32]
    float* w2p  = ws + N + (size_t)2 * N * 64;

    const int B = 256;
    const int ntiles = N / 16;                // 6250
    const int gemm_blocks = 625;              // 10 row-tiles per block

    // norm
    k_init_deg<<<(N + B - 1) / B, B, 0, stream>>>(dinv, N);
    k_deg_edges<<<(E + B - 1) / B, B, 0, stream>>>(dstI, dinv, E);
    k_rsqrt<<<(N + B - 1) / B, B, 0, stream>>>(dinv, N);

    // layer 1: h1 = x @ W1 (WMMA + LDS-staged B), aggregate, bias+relu
    k_gemm_wmma<64><<<gemm_blocks, 128, 0, stream>>>(x, W1, h1, ntiles);
    k_self1<<<(N * 64 + B - 1) / B, B, 0, stream>>>(h1, dinv, agg, N * 64);
    k_edge1<<<((size_t)E * 32 + B - 1) / B, B, 0, stream>>>(srcI, dstI, dinv, h1, agg, E);
    k_bias_relu<<<(N * 64 + B - 1) / B, B, 0, stream>>>(agg, b1, h1, N * 64);

    // layer 2: h2 = h @ W2p (WMMA, padded to 32 cols), aggregate into out
    k_pad_w2<<<(64 * 32 + B - 1) / B, B, 0, stream>>>(W2, w2p, C2);
    k_gemm_wmma<32><<<gemm_blocks, 128, 0, stream>>>(h1, w2p, agg, ntiles);  // agg -> h2
    k_self2<<<((size_t)N * 32 + B - 1) / B, B, 0, stream>>>(agg, dinv, b2, out, N, C2);
    k_edge2<<<((size_t)E * 32 + B - 1) / B, B, 0, stream>>>(srcI, dstI, dinv, agg, out, E, C2);
}